// GCN_47270410060374
// MI455X (gfx1250) — compile-verified
//
#include <hip/hip_runtime.h>

typedef __attribute__((ext_vector_type(2))) float v2f;
typedef __attribute__((ext_vector_type(8))) float v8f;

#define N_NODES_C 100000
#define N_EDGES_C 1600000
#define IN_DIM_C  128
#define HID_DIM_C 64

// ---------------- utility kernels ----------------

__global__ void GCN_zero_kernel(float* __restrict__ p, int n) {
  int i = blockIdx.x * blockDim.x + threadIdx.x;
  if (i < n) p[i] = 0.0f;
}

__global__ void GCN_degree_kernel(const int* __restrict__ src, const int* __restrict__ dst,
                                  float* __restrict__ degO, float* __restrict__ degI, int nE) {
  int e = blockIdx.x * blockDim.x + threadIdx.x;
  if (e < nE) {
    atomicAdd(&degO[src[e]], 1.0f);
    atomicAdd(&degI[dst[e]], 1.0f);
  }
}

__global__ void GCN_rsqrt_kernel(float* __restrict__ d, int n) {
  int i = blockIdx.x * blockDim.x + threadIdx.x;
  if (i < n) d[i] = rsqrtf(fmaxf(d[i], 1.0f));
}

// ---------------- fp32 WMMA GEMM: T = (diag(scale) * A) @ W ----------------
// A: [N, K] row-major, W: [K, 64] row-major, T: [N, 64] row-major.
// One wave (32 lanes) computes a 16x64 output stripe via 4x v8f accumulators.
// V_WMMA_F32_16X16X4_F32 fragment layouts (ISA 7.12.2):
//   A 16x4 f32 (2 VGPRs/lane): lanes 0-15 -> M=lane, K={k0,k0+1};
//                              lanes 16-31 -> M=lane-16, K={k0+2,k0+3}
//   B 4x16 f32 (2 VGPRs/lane): symmetric, lane carries N, half carries K
//   C/D 16x16 f32 (8 VGPRs):   VGPR r -> M=r (lanes 0-15) / M=r+8 (lanes 16-31), N=lane&15
template <int K>
__global__ void GCN_gemm_wmma_kernel(const float* __restrict__ A,
                                     const float* __restrict__ scale,
                                     const float* __restrict__ W,
                                     float* __restrict__ T, int nTiles) {
  const int lane = threadIdx.x & 31;
  const int wave = threadIdx.x >> 5;
  const int tile = blockIdx.x * (blockDim.x >> 5) + wave;
  if (tile >= nTiles) return;  // wave-uniform: EXEC stays all-ones for WMMA

  const int rowBase = tile * 16;
  const int n  = lane & 15;
  const int hi = lane >> 4;
  const int m  = rowBase + n;          // A-fragment row for this lane
  const float s = scale[m];            // fuse D_out^{-1/2} into A load
  const float* __restrict__ Arow = A + (size_t)m * K;

  v8f c0 = {}; v8f c1 = {}; v8f c2 = {}; v8f c3 = {};

#pragma unroll
  for (int k0 = 0; k0 < K; k0 += 4) {
    const int kb = k0 + 2 * hi;
    v2f a;
    a.x = Arow[kb]     * s;
    a.y = Arow[kb + 1] * s;

    const float* __restrict__ w0 = W + (size_t)kb * HID_DIM_C;
    const float* __restrict__ w1 = w0 + HID_DIM_C;
    v2f b0, b1, b2, b3;
    b0.x = w0[n];      b0.y = w1[n];
    b1.x = w0[n + 16]; b1.y = w1[n + 16];
    b2.x = w0[n + 32]; b2.y = w1[n + 32];
    b3.x = w0[n + 48]; b3.y = w1[n + 48];

    c0 = __builtin_amdgcn_wmma_f32_16x16x4_f32(false, a, false, b0, (short)0, c0, false, false);
    c1 = __builtin_amdgcn_wmma_f32_16x16x4_f32(false, a, false, b1, (short)0, c1, false, false);
    c2 = __builtin_amdgcn_wmma_f32_16x16x4_f32(false, a, false, b2, (short)0, c2, false, false);
    c3 = __builtin_amdgcn_wmma_f32_16x16x4_f32(false, a, false, b3, (short)0, c3, false, false);
  }

#pragma unroll
  for (int r = 0; r < 8; ++r) {
    const int row = rowBase + r + 8 * hi;
    float* __restrict__ out = T + (size_t)row * HID_DIM_C;
    out[n]      = c0[r];
    out[n + 16] = c1[r];
    out[n + 32] = c2[r];
    out[n + 48] = c3[r];
  }
}

// ---------------- edge scatter-add: M[dst] += T[src] ----------------
// 16 lanes per edge, float4 per lane -> vectorized gather, f32 atomics into L2.
__global__ void GCN_scatter_kernel(const float* __restrict__ T, const int* __restrict__ src,
                                   const int* __restrict__ dst, float* __restrict__ M, int nE) {
  unsigned tid = blockIdx.x * blockDim.x + threadIdx.x;
  unsigned e = tid >> 4;
  if (e >= (unsigned)nE) return;
  const int cg = (tid & 15u) << 2;
  const int sN = src[e];
  const int dN = dst[e];
  const float4 v = *(const float4*)(T + (size_t)sN * HID_DIM_C + cg);
  float* __restrict__ out = M + (size_t)dN * HID_DIM_C + cg;
  atomicAdd(out + 0, v.x);
  atomicAdd(out + 1, v.y);
  atomicAdd(out + 2, v.z);
  atomicAdd(out + 3, v.w);
}

// ---------------- finalize: out = [relu](M * D_in^{-1/2} + b) ----------------
__global__ void GCN_finalize_kernel(const float* __restrict__ M, const float* __restrict__ invIn,
                                    const float* __restrict__ bias, float* __restrict__ out,
                                    int total, int doRelu) {
  int i = blockIdx.x * blockDim.x + threadIdx.x;
  if (i >= total) return;
  const int node = i >> 6;   // HID_DIM == 64
  const int c    = i & 63;
  float v = M[i] * invIn[node] + bias[c];
  if (doRelu) v = fmaxf(v, 0.0f);
  out[i] = v;
}

// ---------------- driver ----------------

extern "C" void kernel_launch(void* const* d_in, const int* in_sizes, int n_in,
                              void* d_out, int out_size, void* d_ws, size_t ws_size,
                              hipStream_t stream) {
  (void)in_sizes; (void)n_in; (void)out_size; (void)ws_size;

  const float* x  = (const float*)d_in[0];
  const int*   src = (const int*)d_in[1];
  const int*   dst = (const int*)d_in[2];
  const float* W1 = (const float*)d_in[3];
  const float* b1 = (const float*)d_in[4];
  const float* W2 = (const float*)d_in[5];
  const float* b2 = (const float*)d_in[6];
  const float* W3 = (const float*)d_in[7];
  const float* b3 = (const float*)d_in[8];
  float* out = (float*)d_out;

  // Workspace layout (floats): invOut[N] | invIn[N] | T[N*64] | M[N*64] | H[N*64]
  float* ws     = (float*)d_ws;
  float* invOut = ws;
  float* invIn  = invOut + N_NODES_C;
  float* T      = invIn + N_NODES_C;
  float* M      = T + (size_t)N_NODES_C * HID_DIM_C;
  float* H      = M + (size_t)N_NODES_C * HID_DIM_C;

  const int TPB = 256;
  const int nInv = 2 * N_NODES_C;
  const int nOut = N_NODES_C * HID_DIM_C;
  const int nTiles = N_NODES_C / 16;              // 6250, exact
  const int wavesPerBlock = 4;                    // 128 threads = 4 wave32
  const int gemmBlocks = (nTiles + wavesPerBlock - 1) / wavesPerBlock;
  const unsigned scatterThreads = (unsigned)N_EDGES_C * 16u;

  // Degrees -> inv-sqrt (invOut/invIn are contiguous: do both at once)
  GCN_zero_kernel<<<(nInv + TPB - 1) / TPB, TPB, 0, stream>>>(invOut, nInv);
  GCN_degree_kernel<<<(N_EDGES_C + TPB - 1) / TPB, TPB, 0, stream>>>(src, dst, invOut, invIn, N_EDGES_C);
  GCN_rsqrt_kernel<<<(nInv + TPB - 1) / TPB, TPB, 0, stream>>>(invOut, nInv);

  // Layer 1: T = (diag(invOut) * x) @ W1 ; M = scatter(T) ; H = relu(M*invIn + b1)
  GCN_gemm_wmma_kernel<IN_DIM_C><<<gemmBlocks, 32 * wavesPerBlock, 0, stream>>>(x, invOut, W1, T, nTiles);
  GCN_zero_kernel<<<(nOut + TPB - 1) / TPB, TPB, 0, stream>>>(M, nOut);
  GCN_scatter_kernel<<<(scatterThreads + TPB - 1) / TPB, TPB, 0, stream>>>(T, src, dst, M, N_EDGES_C);
  GCN_finalize_kernel<<<(nOut + TPB - 1) / TPB, TPB, 0, stream>>>(M, invIn, b1, H, nOut, 1);

  // Layer 2
  GCN_gemm_wmma_kernel<HID_DIM_C><<<gemmBlocks, 32 * wavesPerBlock, 0, stream>>>(H, invOut, W2, T, nTiles);
  GCN_zero_kernel<<<(nOut + TPB - 1) / TPB, TPB, 0, stream>>>(M, nOut);
  GCN_scatter_kernel<<<(scatterThreads + TPB - 1) / TPB, TPB, 0, stream>>>(T, src, dst, M, N_EDGES_C);
  GCN_finalize_kernel<<<(nOut + TPB - 1) / TPB, TPB, 0, stream>>>(M, invIn, b2, H, nOut, 1);

  // Layer 3 (no ReLU, writes d_out)
  GCN_gemm_wmma_kernel<HID_DIM_C><<<gemmBlocks, 32 * wavesPerBlock, 0, stream>>>(H, invOut, W3, T, nTiles);
  GCN_zero_kernel<<<(nOut + TPB - 1) / TPB, TPB, 0, stream>>>(M, nOut);
  GCN_scatter_kernel<<<(scatterThreads + TPB - 1) / TPB, TPB, 0, stream>>>(T, src, dst, M, N_EDGES_C);
  GCN_finalize_kernel<<<(nOut + TPB - 1) / TPB, TPB, 0, stream>>>(M, invIn, b3, out, nOut, 0);
}